// RobustPolymerGCN_16097537425803
// MI455X (gfx1250) — compile-verified
//
#include <hip/hip_runtime.h>
#include <math.h>

#define N_NODES 200000
#define N_EDGES 800000
#define FIN     32
#define HID     128
#define N_GRAPH 8192
#define N_OUT   5
#define BN_EPS  1e-5f

typedef float v2f __attribute__((ext_vector_type(2)));
typedef float v8f __attribute__((ext_vector_type(8)));

__device__ __forceinline__ void atomicAddF32(float* p, float v) {
    __hip_atomic_fetch_add(p, v, __ATOMIC_RELAXED, __HIP_MEMORY_SCOPE_AGENT);
}

// ---------------------------------------------------------------------------
// GEMM  Y[n,HID] = X[n,K] @ W[K,HID]   via V_WMMA_F32_16X16X4_F32 (fp32 path)
// block = 256 threads (8 wave32); block computes a 16-row x 128-col slab.
// A-tile (16xK) staged in LDS with padded stride (conflict-free lane reads);
// each wave owns one 16x16 tile, K/4 chained WMMAs, acc resident in 8 VGPRs.
// ---------------------------------------------------------------------------
template <int K>
__global__ __launch_bounds__(256) void gemm_wmma_f32(const float* __restrict__ X,
                                                     const float* __restrict__ W,
                                                     float* __restrict__ Y) {
    constexpr int LDK = K + 4;                  // pad: avoid 16-way LDS bank conflicts
    __shared__ float lds[16 * LDK];
    const int rowBase = blockIdx.x * 16;
    const int tid = threadIdx.x;

    // cooperative coalesced load of X[rowBase:rowBase+16, 0:K]
    for (int i = tid; i < 16 * K; i += 256) {
        const int rr = i / K, cc = i - rr * K;
        lds[rr * LDK + cc] = X[(size_t)(rowBase + rr) * K + cc];
    }
    __syncthreads();

    const int wave    = tid >> 5;               // 0..7 -> column tile
    const int lane    = tid & 31;
    const int colBase = wave * 16;
    const int m       = lane & 15;              // M index (A) / N index (B)
    const int koff    = (lane >> 4) << 1;       // lanes 16..31 hold K+2,K+3

    v8f acc = {};
#pragma unroll
    for (int k0 = 0; k0 < K; k0 += 4) {
        v2f a, b;
        // A 16x4 frag: VGPR0=K(koff), VGPR1=K(koff+1); M striped over lanes 0-15/16-31
        a.x = lds[m * LDK + k0 + koff];
        a.y = lds[m * LDK + k0 + koff + 1];
        // B 4x16 frag: row K striped across lanes within each VGPR
        b.x = W[(size_t)(k0 + koff) * HID + colBase + m];
        b.y = W[(size_t)(k0 + koff + 1) * HID + colBase + m];
        acc = __builtin_amdgcn_wmma_f32_16x16x4_f32(false, a, false, b,
                                                    (short)0, acc, false, false);
    }

    // C/D layout: VGPR i -> M = i (lanes 0-15) / i+8 (lanes 16-31), N = lane&15
    const int rBase = rowBase + ((lane >> 4) << 3);
    const int c     = colBase + m;
#pragma unroll
    for (int i = 0; i < 8; ++i)
        Y[(size_t)(rBase + i) * HID + c] = acc[i];
}

// ---------------------------------------------------------------------------
// Degree / normalization:  deg[c] = 1 + #incoming edges;  dinv = rsqrt(deg)
// ---------------------------------------------------------------------------
__global__ void deg_init_kernel(float* __restrict__ d, int n) {
    int i = blockIdx.x * blockDim.x + threadIdx.x;
    if (i < n) d[i] = 1.0f;                     // self-loop
}
__global__ void deg_accum_kernel(const int* __restrict__ col, float* __restrict__ d, int e) {
    int i = blockIdx.x * blockDim.x + threadIdx.x;
    if (i < e) atomicAddF32(&d[col[i]], 1.0f);
}
__global__ void deg_inv_kernel(float* __restrict__ d, int n) {
    int i = blockIdx.x * blockDim.x + threadIdx.x;
    if (i < n) d[i] = rsqrtf(d[i]);             // deg >= 1 always
}

// ---------------------------------------------------------------------------
// Aggregation: P[c] = dinv[c]^2 * Q[c]  (self loop)  +  sum_e dinv[r]dinv[c] Q[r]
// Gathers/atomics resolve in the 192MB L2 (whole 102MB feature matrix resident).
// ---------------------------------------------------------------------------
__global__ void agg_selfloop_kernel(const float* __restrict__ Q,
                                    const float* __restrict__ dinv,
                                    float* __restrict__ P, int total) {
    int t = blockIdx.x * blockDim.x + threadIdx.x;
    if (t >= total) return;
    const int node = t >> 7;                    // /HID
    const float d = dinv[node];
    P[t] = d * d * Q[t];
}

__global__ void agg_edges_kernel(const int* __restrict__ row, const int* __restrict__ col,
                                 const float* __restrict__ dinv,
                                 const float* __restrict__ Q, float* __restrict__ P, int e) {
    int t = blockIdx.x * blockDim.x + threadIdx.x;
    const int edge = t >> 5;                    // one wave32 per edge
    if (edge >= e) return;
    const int lane = t & 31;
    const int r = row[edge], c = col[edge];
    const float w = dinv[r] * dinv[c];
    const float* __restrict__ src = Q + (size_t)r * HID;
    float*       __restrict__ dst = P + (size_t)c * HID;
#pragma unroll
    for (int j = lane; j < HID; j += 32)
        atomicAddF32(&dst[j], w * src[j]);
}

__global__ void bias_bn_relu_kernel(float* __restrict__ h,
                                    const float* __restrict__ bias,
                                    const float* __restrict__ gamma,
                                    const float* __restrict__ beta,
                                    const float* __restrict__ mean,
                                    const float* __restrict__ var, int total) {
    int t = blockIdx.x * blockDim.x + threadIdx.x;
    if (t >= total) return;
    const int j = t & (HID - 1);
    float v = h[t] + bias[j];
    v = (v - mean[j]) * rsqrtf(var[j] + BN_EPS) * gamma[j] + beta[j];
    h[t] = fmaxf(v, 0.0f);
}

// ---------------------------------------------------------------------------
// Global mean pool + linear head
// ---------------------------------------------------------------------------
__global__ void zero_kernel(float* __restrict__ p, int n) {
    int i = blockIdx.x * blockDim.x + threadIdx.x;
    if (i < n) p[i] = 0.0f;
}

__global__ void pool_accum_kernel(const float* __restrict__ h,
                                  const int* __restrict__ batch,
                                  float* __restrict__ gsum, float* __restrict__ gcnt, int n) {
    int t = blockIdx.x * blockDim.x + threadIdx.x;
    const int node = t >> 5;
    if (node >= n) return;
    const int lane = t & 31;
    const int g = batch[node];
    const float* __restrict__ src = h + (size_t)node * HID;
    float*       __restrict__ dst = gsum + (size_t)g * HID;
#pragma unroll
    for (int j = lane; j < HID; j += 32)
        atomicAddF32(&dst[j], src[j]);
    if (lane == 0) atomicAddF32(&gcnt[g], 1.0f);
}

__global__ void readout_kernel(const float* __restrict__ gsum,
                               const float* __restrict__ gcnt,
                               const float* __restrict__ Wout,
                               const float* __restrict__ bout,
                               float* __restrict__ out) {
    int t = blockIdx.x * blockDim.x + threadIdx.x;
    if (t >= N_GRAPH * N_OUT) return;
    const int g = t / N_OUT, o = t - g * N_OUT;
    const float inv = 1.0f / fmaxf(gcnt[g], 1.0f);
    const float* __restrict__ gs = gsum + (size_t)g * HID;
    float s = 0.0f;
#pragma unroll 8
    for (int j = 0; j < HID; ++j)
        s = fmaf(gs[j], Wout[j * N_OUT + o], s);
    out[t] = s * inv + bout[o];
}

// ---------------------------------------------------------------------------
extern "C" void kernel_launch(void* const* d_in, const int* in_sizes, int n_in,
                              void* d_out, int out_size, void* d_ws, size_t ws_size,
                              hipStream_t stream) {
    (void)in_sizes; (void)n_in; (void)out_size; (void)ws_size;

    const float* x     = (const float*)d_in[0];
    const int*   eidx  = (const int*)d_in[1];
    const int*   batch = (const int*)d_in[2];
    const float* W[3]     = {(const float*)d_in[3],  (const float*)d_in[9],  (const float*)d_in[15]};
    const float* b[3]     = {(const float*)d_in[4],  (const float*)d_in[10], (const float*)d_in[16]};
    const float* gamma[3] = {(const float*)d_in[5],  (const float*)d_in[11], (const float*)d_in[17]};
    const float* beta[3]  = {(const float*)d_in[6],  (const float*)d_in[12], (const float*)d_in[18]};
    const float* mean[3]  = {(const float*)d_in[7],  (const float*)d_in[13], (const float*)d_in[19]};
    const float* var[3]   = {(const float*)d_in[8],  (const float*)d_in[14], (const float*)d_in[20]};
    const float* Wout = (const float*)d_in[21];
    const float* bout = (const float*)d_in[22];
    const int* row = eidx;              // edge_index[0]
    const int* col = eidx + N_EDGES;    // edge_index[1]
    float* out = (float*)d_out;

    // workspace layout (floats): P | Q | dinv | gsum | gcnt  (~210 MB)
    const size_t NH = (size_t)N_NODES * HID;
    float* P    = (float*)d_ws;
    float* Q    = P + NH;
    float* dinv = Q + NH;
    float* gsum = dinv + N_NODES;
    float* gcnt = gsum + (size_t)N_GRAPH * HID;

    const int TB = 256;
    const int gNodes = (N_NODES + TB - 1) / TB;
    const int gEdges = (N_EDGES + TB - 1) / TB;
    const int gNH    = (int)(NH / TB);                 // 25.6M / 256 = 100000
    const int gEdge32 = (int)(((size_t)N_EDGES * 32) / TB);
    const int gNode32 = (int)(((size_t)N_NODES * 32) / TB);
    const int gGemm   = N_NODES / 16;                  // 12500

    // --- degree normalization (recomputed every call: deterministic) ---
    deg_init_kernel<<<gNodes, TB, 0, stream>>>(dinv, N_NODES);
    deg_accum_kernel<<<gEdges, TB, 0, stream>>>(col, dinv, N_EDGES);
    deg_inv_kernel<<<gNodes, TB, 0, stream>>>(dinv, N_NODES);

    // --- layer 0 (K = FIN = 32) ---
    gemm_wmma_f32<FIN><<<gGemm, TB, 0, stream>>>(x, W[0], Q);
    agg_selfloop_kernel<<<gNH, TB, 0, stream>>>(Q, dinv, P, (int)NH);
    agg_edges_kernel<<<gEdge32, TB, 0, stream>>>(row, col, dinv, Q, P, N_EDGES);
    bias_bn_relu_kernel<<<gNH, TB, 0, stream>>>(P, b[0], gamma[0], beta[0], mean[0], var[0], (int)NH);

    // --- layers 1,2 (K = HID = 128) ---
    for (int l = 1; l < 3; ++l) {
        gemm_wmma_f32<HID><<<gGemm, TB, 0, stream>>>(P, W[l], Q);
        agg_selfloop_kernel<<<gNH, TB, 0, stream>>>(Q, dinv, P, (int)NH);
        agg_edges_kernel<<<gEdge32, TB, 0, stream>>>(row, col, dinv, Q, P, N_EDGES);
        bias_bn_relu_kernel<<<gNH, TB, 0, stream>>>(P, b[l], gamma[l], beta[l], mean[l], var[l], (int)NH);
    }

    // --- global mean pool + head ---
    const int poolN = N_GRAPH * HID + N_GRAPH;
    zero_kernel<<<(poolN + TB - 1) / TB, TB, 0, stream>>>(gsum, poolN);  // gcnt adjacent
    pool_accum_kernel<<<gNode32, TB, 0, stream>>>(P, batch, gsum, gcnt, N_NODES);
    readout_kernel<<<(N_GRAPH * N_OUT + TB - 1) / TB, TB, 0, stream>>>(gsum, gcnt, Wout, bout, out);
}